// GateLinearAttention_45526653338169
// MI455X (gfx1250) — compile-verified
//
#include <hip/hip_runtime.h>
#include <math.h>

// ---------------------------------------------------------------------------
// GLA forward for MI455X (gfx1250, wave32).
//   B=2, T=512, DIM=1024, H=8, HK=HV=128, GLR=16.
// Pipeline:
//   1) q,k,v,g projections: fp32 WMMA GEMM (v_wmma_f32_16x16x4_f32), LDS fed
//      by a double-buffered Tensor Data Mover pipeline (tensor_load_to_lds,
//      s_wait_tensorcnt 2 ping-pong)
//   2) low-rank gate:  gk = logsigmoid((x@Wgk1)@Wgk2 + b)/16
//   3) recurrence: S_t = diag(exp(gk)) S_{t-1} + k^T v ; o = q S  (reg-resident)
//   4) gated RMSNorm with swish gate
//   5) out = o_norm @ Wo : same WMMA GEMM
// ---------------------------------------------------------------------------

#define DIMX 1024
#define HX   8
#define HKX  128
#define HVX  128
#define GLRX 16
#define TX   512
#define BX   2
#define NTOK (BX * TX)          // 1024 token rows

typedef __attribute__((ext_vector_type(2))) float    v2f;
typedef __attribute__((ext_vector_type(8))) float    v8f;
typedef __attribute__((ext_vector_type(4))) uint32_t u32x4;
typedef __attribute__((ext_vector_type(4))) int32_t  i32x4;
typedef __attribute__((ext_vector_type(8))) int32_t  i32x8;

#if __has_builtin(__builtin_amdgcn_tensor_load_to_lds)
#define USE_TDM 1
#else
#define USE_TDM 0
#endif

// ---------------------------------------------------------------------------
// Issue a 2-D TDM tile load (global -> LDS).  All inputs are wave-uniform.
//   lds_off  : byte offset of destination inside the WG's LDS allocation
//   gptr     : global address of tile start
//   td0/td1  : tensor dims (elements, 4B each); str0: row stride (elements)
//   tile0/1  : tile dims (elements / rows)
//   pad_*    : TDM LDS padding (creates bank-skewed LDS strides in the DMA)
// ---------------------------------------------------------------------------
#if USE_TDM
__device__ __forceinline__ void tdm_load_2d(uint32_t lds_off, const void* gptr,
                                            uint32_t td0, uint32_t td1,
                                            uint32_t str0,
                                            uint32_t tile0, uint32_t tile1,
                                            uint32_t pad_en, uint32_t pad_iv,
                                            uint32_t pad_amt) {
  uint64_t ga = (uint64_t)(uintptr_t)gptr;
  u32x4 g0;
  g0[0] = 1u;                                    // count=1 (valid user D#)
  g0[1] = lds_off;                               // lds_addr
  g0[2] = (uint32_t)ga;                          // global_addr[31:0]
  g0[3] = (uint32_t)(ga >> 32) | 0x80000000u;    // global_addr[56:32] | type=2
  i32x8 g1;
  g1[0] = (int32_t)((2u << 16) |                 // data_size = 4B
                    (pad_en << 20) | (pad_iv << 22) | (pad_amt << 25));
  g1[1] = (int32_t)((td0 & 0xFFFFu) << 16);                      // dim0 lo
  g1[2] = (int32_t)((td0 >> 16) | ((td1 & 0xFFFFu) << 16));      // dim0 hi|dim1 lo
  g1[3] = (int32_t)((td1 >> 16) | (tile0 << 16));                // dim1 hi|tile0
  g1[4] = (int32_t)(tile1 & 0xFFFFu);                            // tile1, tile2=0
  g1[5] = (int32_t)str0;                                         // dim0_stride lo
  g1[6] = 0;                                                     // stride hi / d1s
  g1[7] = 0;
  i32x4 z4 = {0, 0, 0, 0};
#if defined(__clang_major__) && __clang_major__ >= 23
  i32x8 z8 = {0, 0, 0, 0, 0, 0, 0, 0};
  __builtin_amdgcn_tensor_load_to_lds(g0, g1, z4, z4, z8, 0);
#else
  __builtin_amdgcn_tensor_load_to_lds(g0, g1, z4, z4, 0);
#endif
}
#endif

// ---------------------------------------------------------------------------
// fp32 WMMA GEMM:  C[M,N] = alpha * A[M,K] @ B[K,N]   (row-major)
// Block: 256 threads = 8 waves (2x4) -> 32(M) x 128(N) tile; each wave keeps
// two 16x16 accumulators (cols n and n+64) so the A fragment is reused.
// K staged through LDS in KT=32 steps by a DOUBLE-BUFFERED TDM pipeline:
// tile i+1's DMA is issued before waiting tensorcnt<=2 for tile i, so the
// Tensor Data Mover runs concurrently with the 16-WMMA burst.
// TDM padding gives As stride-33 and Bs stride-136 bank skew.
// ---------------------------------------------------------------------------
#define KT 32

__global__ void __launch_bounds__(256)
gla_gemm_wmma_f32(const float* __restrict__ A, const float* __restrict__ B,
                  float* __restrict__ C, int M, int Nn, int K, float alpha) {
  __shared__ float As[2][32][KT + 1];  // stride 33 (TDM pad 1 dw / 32 dw)
  __shared__ float Bs[2][KT][136];     // stride 136 (TDM pad 8 dw / 128 dw)

  const int lane = threadIdx.x & 31;
  const int wave = threadIdx.x >> 5;     // 0..7
  const int wm   = wave >> 2;            // 0..1 : 16-row group in block
  const int wn   = wave & 3;             // 0..3 : 16-col group in block
  const int m0   = blockIdx.y * 32;
  const int n0   = blockIdx.x * 128;

  v8f acc0 = {};
  v8f acc1 = {};

  const int mrow = (wm << 4) + (lane & 15);     // A row inside block tile
  const int nc0  = (wn << 4) + (lane & 15);     // first B col
  const int nc1  = nc0 + 64;                    // second B col
  const int koff = (lane < 16) ? 0 : 2;         // K pair selected by half-wave

  const int ntiles = K / KT;

#if USE_TDM
  // Preload tile 0 into buffer 0 (wave 0 drives the DMA engine).
  if (threadIdx.x < 32) {
    tdm_load_2d((uint32_t)(uintptr_t)(void*)&As[0][0][0],
                &A[(size_t)m0 * K], (uint32_t)K, (uint32_t)M, (uint32_t)K,
                KT, 32, 1u, 4u, 0u);
    tdm_load_2d((uint32_t)(uintptr_t)(void*)&Bs[0][0][0],
                &B[(size_t)0 * Nn + n0], (uint32_t)Nn, (uint32_t)K,
                (uint32_t)Nn, 128, KT, 1u, 6u, 7u);
  }
#endif

  for (int it = 0; it < ntiles; ++it) {
    const int cur = it & 1;
#if USE_TDM
    if (threadIdx.x < 32) {
      if (it + 1 < ntiles) {
        const int nk = (it + 1) * KT;
        const int nxt = cur ^ 1;
        // A tile: 32 rows x KT, pad 1 dword per 32 -> LDS stride 33
        tdm_load_2d((uint32_t)(uintptr_t)(void*)&As[nxt][0][0],
                    &A[(size_t)m0 * K + nk], (uint32_t)K, (uint32_t)M,
                    (uint32_t)K, KT, 32, 1u, 4u, 0u);
        // B tile: KT rows x 128, pad 8 dwords per 128 -> LDS stride 136
        tdm_load_2d((uint32_t)(uintptr_t)(void*)&Bs[nxt][0][0],
                    &B[(size_t)nk * Nn + n0], (uint32_t)Nn, (uint32_t)K,
                    (uint32_t)Nn, 128, KT, 1u, 6u, 7u);
        // tile `it`'s 2 loads retired when <=2 younger remain (in-order)
        __builtin_amdgcn_s_wait_tensorcnt(2);
      } else {
        __builtin_amdgcn_s_wait_tensorcnt(0);
      }
    }
    __syncthreads();   // tile `cur` visible to all waves
#else
    __syncthreads();   // previous compute done before overwrite
    {
      const int k0 = it * KT;
      for (int idx = threadIdx.x; idx < 32 * KT; idx += 256) {
        int r = idx >> 5, cc = idx & (KT - 1);
        As[cur][r][cc] = A[(size_t)(m0 + r) * K + k0 + cc];
      }
      for (int idx = threadIdx.x; idx < KT * 128; idx += 256) {
        int r = idx >> 7, cc = idx & 127;
        Bs[cur][r][cc] = B[(size_t)(k0 + r) * Nn + n0 + cc];
      }
    }
    __syncthreads();
#endif

#pragma unroll
    for (int kk = 0; kk < KT; kk += 4) {
      // A 16x4 layout: VGPR0 = K (0|2), VGPR1 = K (1|3), halves by lane<16
      v2f a, b0, b1;
      a.x  = As[cur][mrow][kk + koff];
      a.y  = As[cur][mrow][kk + koff + 1];
      b0.x = Bs[cur][kk + koff][nc0];
      b0.y = Bs[cur][kk + koff + 1][nc0];
      b1.x = Bs[cur][kk + koff][nc1];
      b1.y = Bs[cur][kk + koff + 1][nc1];
      acc0 = __builtin_amdgcn_wmma_f32_16x16x4_f32(
          false, a, false, b0, (short)0, acc0, false, false);
      acc1 = __builtin_amdgcn_wmma_f32_16x16x4_f32(
          false, a, false, b1, (short)0, acc1, false, false);
    }
    __syncthreads();   // all waves done with buf `cur` before its next DMA
  }

  // C/D layout: VGPR r -> M = r (+8 for lanes 16..31), N = lane&15
  const int row0 = m0 + (wm << 4) + ((lane < 16) ? 0 : 8);
  const int colA = n0 + (wn << 4) + (lane & 15);
  const int colB = colA + 64;
#pragma unroll
  for (int r = 0; r < 8; ++r) {
    C[(size_t)(row0 + r) * Nn + colA] = alpha * acc0[r];
    C[(size_t)(row0 + r) * Nn + colB] = alpha * acc1[r];
  }
}

// ---------------------------------------------------------------------------
// Low-rank gate stage 1: tmp[N,16] = x[N,1024] @ Wgk1[1024,16]
// ---------------------------------------------------------------------------
__global__ void __launch_bounds__(256)
gla_lowrank1(const float* __restrict__ x, const float* __restrict__ Wgk1,
             float* __restrict__ tmp) {
  int idx = blockIdx.x * 256 + threadIdx.x;      // NTOK*GLRX threads
  int i = idx >> 4, r = idx & 15;
  float acc = 0.f;
  for (int kx = 0; kx < DIMX; ++kx)
    acc += x[i * DIMX + kx] * Wgk1[kx * GLRX + r];
  tmp[idx] = acc;
}

// ---------------------------------------------------------------------------
// Low-rank gate stage 2 + logsigmoid:  gk = logsigmoid(tmp @ Wgk2 + b) / 16
// ---------------------------------------------------------------------------
__global__ void __launch_bounds__(256)
gla_lowrank2_gate(const float* __restrict__ tmp, const float* __restrict__ Wgk2,
                  const float* __restrict__ bgk2, float* __restrict__ gk) {
  int idx = blockIdx.x * 256 + threadIdx.x;      // NTOK*DIMX threads
  int i = idx >> 10, d = idx & (DIMX - 1);
  float acc = bgk2[d];
#pragma unroll
  for (int r = 0; r < GLRX; ++r)
    acc += tmp[(i << 4) + r] * Wgk2[r * DIMX + d];
  float z  = acc;
  float ls = (z < 0.f) ? (z - log1pf(__expf(z))) : (-log1pf(__expf(-z)));
  gk[idx] = ls * (1.0f / 16.0f);
}

// ---------------------------------------------------------------------------
// Decay-gated recurrence. One workgroup per (b, h, 32-col v-block): 64 WGs.
// 256 threads: thread owns v-column c = tid&31 of the block and a 16-row
// k-strip (strip = tid>>5), keeping 16 floats of the 128x128 state in VGPRs.
// ---------------------------------------------------------------------------
__global__ void __launch_bounds__(256)
gla_recurrence(const float* __restrict__ q, const float* __restrict__ k,
               const float* __restrict__ v, const float* __restrict__ gk,
               float* __restrict__ o) {
  const int bh   = blockIdx.x >> 2;     // 0..BX*HX-1
  const int vblk = blockIdx.x & 3;      // which 32 v-columns
  const int b  = bh / HX;
  const int h  = bh % HX;
  const int tid   = threadIdx.x;
  const int c     = tid & 31;           // v column within block
  const int strip = tid >> 5;           // 0..7 -> k rows [strip*16, +16)
  const int kb    = strip * 16;

  __shared__ float qs[HKX], ks[HKX], egs[HKX], vs[32], partial[256];

  float S[16];
#pragma unroll
  for (int j = 0; j < 16; ++j) S[j] = 0.f;

  for (int t = 0; t < TX; ++t) {
    const int base = ((b * TX + t) * DIMX) + h * HKX;
    if (tid < HKX) {
      qs[tid]  = q[base + tid];
      ks[tid]  = k[base + tid];
      egs[tid] = __expf(gk[base + tid]);
    } else if (tid < HKX + 32) {
      vs[tid - HKX] = v[base + vblk * 32 + (tid - HKX)];
    }
    __syncthreads();

    const float vc = vs[c];
    float oacc = 0.f;
#pragma unroll
    for (int j = 0; j < 16; ++j) {
      const int kk = kb + j;
      float s = S[j] * egs[kk] + ks[kk] * vc;   // diag(exp(gk)) S + k^T v
      S[j] = s;
      oacc += qs[kk] * s;                        // q . S  (strip partial)
    }
    partial[tid] = oacc;
    __syncthreads();

    if (tid < 32) {
      float r = 0.f;
#pragma unroll
      for (int s2 = 0; s2 < 8; ++s2) r += partial[s2 * 32 + tid];
      o[base + vblk * 32 + tid] = r;
    }
    __syncthreads();   // protect LDS before next step's overwrite
  }
}

// ---------------------------------------------------------------------------
// Gated RMSNorm + swish gate: one block per (token, head), 128 threads.
// ---------------------------------------------------------------------------
__global__ void __launch_bounds__(128)
gla_norm_gate(const float* __restrict__ o_pre, const float* __restrict__ g,
              const float* __restrict__ w, float* __restrict__ o_out) {
  const int bid = blockIdx.x;           // NTOK*HX blocks
  const int i = bid / HX, h = bid % HX;
  const int c = threadIdx.x;
  const int off = i * DIMX + h * HVX + c;

  __shared__ float red[128];
  float val = o_pre[off];
  red[c] = val * val;
  __syncthreads();
#pragma unroll
  for (int s = 64; s > 0; s >>= 1) {
    if (c < s) red[c] += red[c + s];
    __syncthreads();
  }
  const float rms = rsqrtf(red[0] * (1.0f / HVX) + 1e-5f);
  const float gv  = g[off];
  const float sw  = gv * (1.f / (1.f + __expf(-gv)));
  o_out[off] = val * rms * w[c] * sw;
}

// ---------------------------------------------------------------------------
extern "C" void kernel_launch(void* const* d_in, const int* in_sizes, int n_in,
                              void* d_out, int out_size, void* d_ws, size_t ws_size,
                              hipStream_t stream) {
  const float* x    = (const float*)d_in[0];
  const float* Wq   = (const float*)d_in[1];
  const float* Wk   = (const float*)d_in[2];
  const float* Wv   = (const float*)d_in[3];
  const float* Wg   = (const float*)d_in[4];
  const float* Wgk1 = (const float*)d_in[5];
  const float* Wgk2 = (const float*)d_in[6];
  const float* bgk2 = (const float*)d_in[7];
  const float* gnw  = (const float*)d_in[8];
  const float* Wo   = (const float*)d_in[9];
  float* out = (float*)d_out;

  const size_t ND = (size_t)NTOK * DIMX;
  float* ws    = (float*)d_ws;
  float* q     = ws + 0 * ND;
  float* k     = ws + 1 * ND;
  float* v     = ws + 2 * ND;
  float* g     = ws + 3 * ND;
  float* gk    = ws + 4 * ND;
  float* o_pre = ws + 5 * ND;
  float* o_nrm = ws + 6 * ND;
  float* tmpLR = ws + 7 * ND;     // NTOK*GLRX floats

  dim3 gemmGrid(DIMX / 128, NTOK / 32);
  const float qscale = 0.08838834764831845f;   // HK^-0.5

  // 1) projections (fp32 WMMA, double-buffered TDM-fed LDS)
  gla_gemm_wmma_f32<<<gemmGrid, 256, 0, stream>>>(x, Wq, q, NTOK, DIMX, DIMX, qscale);
  gla_gemm_wmma_f32<<<gemmGrid, 256, 0, stream>>>(x, Wk, k, NTOK, DIMX, DIMX, 1.0f);
  gla_gemm_wmma_f32<<<gemmGrid, 256, 0, stream>>>(x, Wv, v, NTOK, DIMX, DIMX, 1.0f);
  gla_gemm_wmma_f32<<<gemmGrid, 256, 0, stream>>>(x, Wg, g, NTOK, DIMX, DIMX, 1.0f);

  // 2) low-rank forget gate
  gla_lowrank1<<<(NTOK * GLRX) / 256, 256, 0, stream>>>(x, Wgk1, tmpLR);
  gla_lowrank2_gate<<<(NTOK * DIMX) / 256, 256, 0, stream>>>(tmpLR, Wgk2, bgk2, gk);

  // 3) gated linear-attention recurrence (register-resident state, 64 WGs)
  gla_recurrence<<<BX * HX * 4, 256, 0, stream>>>(q, k, v, gk, o_pre);

  // 4) gated RMSNorm + swish
  gla_norm_gate<<<NTOK * HX, 128, 0, stream>>>(o_pre, g, gnw, o_nrm);

  // 5) output projection (fp32 WMMA)
  gla_gemm_wmma_f32<<<gemmGrid, 256, 0, stream>>>(o_nrm, Wo, out, NTOK, DIMX, DIMX, 1.0f);
}